// SpGQAFlashDecodeAttentionIrisFusedFull_45664092291683
// MI455X (gfx1250) — compile-verified
//
#include <hip/hip_runtime.h>

typedef float v2f __attribute__((ext_vector_type(2)));
typedef float v8f __attribute__((ext_vector_type(8)));

#define BQ   16
#define HQ   32
#define DH   128
#define TMAX 4096
#define KVHN 8
#define GRP  4
#define NW   16                    // waves per block (wave32)
#define NTHREADS (NW * 32)
#define QSTR 132                   // padded LDS row stride (floats)
#define SCALE_F 0.08838834764831845f
#define NEG_BIG (-1.0e30f)

__device__ __forceinline__ v8f vzero8() {
  v8f v;
#pragma unroll
  for (int i = 0; i < 8; ++i) v[i] = 0.0f;
  return v;
}

__device__ __forceinline__ v8f wmma_f32(v2f a, v2f b, v8f c) {
  // V_WMMA_F32_16X16X4_F32 : D = A(16x4) * B(4x16) + C(16x16)
  return __builtin_amdgcn_wmma_f32_16x16x4_f32(false, a, false, b, (short)0, c,
                                               false, false);
}

__global__ __launch_bounds__(NTHREADS, 1)
void gqa_flash_decode_kernel(const float* __restrict__ q,
                             const float* __restrict__ kc,
                             const float* __restrict__ vc,
                             const int*   __restrict__ kvlens,
                             const int*   __restrict__ btab,
                             float*       __restrict__ out) {
  __shared__ float qbuf[16 * QSTR];        // Q^T staging, shared by all waves
  __shared__ float cacc[NW][GRP * DH];     // per-wave partial O
  __shared__ float wmS[NW * GRP];
  __shared__ float wlS[NW * GRP];

  const int b    = blockIdx.x >> 3;
  const int kvh  = blockIdx.x & 7;
  const int tid  = threadIdx.x;
  const int wave = tid >> 5;
  const int lane = tid & 31;
  const int half = lane >> 4;   // 0: lanes 0-15, 1: lanes 16-31
  const int l16  = lane & 15;

  const int kvlen = kvlens[b];

  // ---- stage Q into LDS (rows >= GRP stay zero) ----------------------------
  for (int idx = tid; idx < 16 * QSTR; idx += NTHREADS) qbuf[idx] = 0.0f;
  __syncthreads();
  {
    const int g = tid >> 7;      // NTHREADS == 512 == GRP*DH
    const int d = tid & 127;
    qbuf[g * QSTR + d] = q[((size_t)b * HQ + (size_t)(kvh * GRP + g)) * DH + d];
  }
  __syncthreads();

  v8f oacc[8];
#pragma unroll
  for (int dt = 0; dt < 8; ++dt) oacc[dt] = vzero8();

  float m = NEG_BIG, l = 0.0f;   // per-lane stats for column g = l16

  const int qoff = l16 * QSTR + 2 * half;

  for (int tbase = wave * 16; tbase < kvlen; tbase += NW * 16) {
    // page for kv position t = tbase + l16 (same in both halves)
    const int pagev = btab[b * TMAX + tbase + l16];

    // ---- prefetch next tile's K/V rows into cache ---------------------------
    const int ntb = tbase + NW * 16;
    if (ntb < kvlen) {
      const int np = btab[b * TMAX + ntb + l16];
      const float* kp = kc + ((size_t)np * KVHN + kvh) * DH + half * 64;
      const float* vp = vc + ((size_t)np * KVHN + kvh) * DH + half * 64;
#pragma unroll
      for (int i = 0; i < 4; ++i) {
        __builtin_prefetch(kp + i * 16, 0, 3);
        __builtin_prefetch(vp + i * 16, 0, 3);
      }
    }

    // ---- S^T = K * Q^T  (M=16 t-rows, N=16 g-cols, K=128 in 32 steps) ------
    // A-frag (K): lane holds row M=t=l16, elems d = 4s+2*half {+0,+1}
    // B-frag (Q): lane holds col N=g=l16, elems d = 4s+2*half {+0,+1} (from LDS)
    const float* krow = kc + ((size_t)pagev * KVHN + kvh) * DH + 2 * half;
    v8f s0 = vzero8(), s1 = vzero8();
#pragma unroll
    for (int cc = 0; cc < 4; ++cc) {
      v2f kf[8], qv[8];
#pragma unroll
      for (int i = 0; i < 8; ++i)
        kf[i] = __builtin_nontemporal_load(
            reinterpret_cast<const v2f*>(krow + 4 * (cc * 8 + i)));
#pragma unroll
      for (int i = 0; i < 8; ++i)
        qv[i] = *reinterpret_cast<const v2f*>(&qbuf[qoff + 4 * (cc * 8 + i)]);
#pragma unroll
      for (int i = 0; i < 8; i += 2) {
        s0 = wmma_f32(kf[i],     qv[i],     s0);
        s1 = wmma_f32(kf[i + 1], qv[i + 1], s1);
      }
    }
    const v8f sacc = s0 + s1;

    // ---- online softmax: element j <-> t = tbase + j + 8*half; col g = l16 -
    float sj[8];
#pragma unroll
    for (int j = 0; j < 8; ++j) {
      const bool v = (tbase + j + 8 * half) < kvlen;
      sj[j] = v ? (sacc[j] * SCALE_F) : NEG_BIG;
    }
    float tm = sj[0];
#pragma unroll
    for (int j = 1; j < 8; ++j) tm = fmaxf(tm, sj[j]);
    tm = fmaxf(tm, __shfl_xor(tm, 16));          // fold both halves of column
    const float nm = fmaxf(m, tm);
    const float al = __expf(m - nm);
    m = nm;
    float pc[8];
    float ps = 0.0f;
#pragma unroll
    for (int j = 0; j < 8; ++j) {
      pc[j] = __expf(sj[j] - nm);
      ps += pc[j];
    }
    ps += __shfl_xor(ps, 16);
    l = l * al + ps;
#pragma unroll
    for (int dt = 0; dt < 8; ++dt) oacc[dt] *= al;

    // ---- build P^T B-fragments from C-layout pc via shuffles ---------------
    // value P(t,g) lives at vgpr (t&7), lane (g + 16*(t>>3))
    v2f pf[4];
#pragma unroll
    for (int u = 0; u < 4; ++u) {
      const int sl = l16 + ((u & 2) ? 16 : 0);   // t>=8 for u>=2
      const int bj = 4 * (u & 1);
      const float a0 = __shfl(pc[bj + 0], sl);   // t = 4u   (lower half)
      const float a1 = __shfl(pc[bj + 1], sl);   // t = 4u+1 (lower half)
      const float a2 = __shfl(pc[bj + 2], sl);   // t = 4u+2 (upper half)
      const float a3 = __shfl(pc[bj + 3], sl);   // t = 4u+3 (upper half)
      pf[u].x = half ? a2 : a0;
      pf[u].y = half ? a3 : a1;
    }

    // ---- O^T += V^T * P^T  (M=16 d-rows per d-tile, N=16 g-cols, K=16 t) ---
    const float* vb0[4];
    const float* vb1[4];
#pragma unroll
    for (int u = 0; u < 4; ++u) {
      const int p0 = __shfl(pagev, 4 * u + 2 * half);
      const int p1 = __shfl(pagev, 4 * u + 1 + 2 * half);
      vb0[u] = vc + ((size_t)p0 * KVHN + kvh) * DH + l16;
      vb1[u] = vc + ((size_t)p1 * KVHN + kvh) * DH + l16;
    }
#pragma unroll
    for (int dt = 0; dt < 8; ++dt) {
      v8f c = oacc[dt];
#pragma unroll
      for (int u = 0; u < 4; ++u) {
        v2f vf;
        vf.x = __builtin_nontemporal_load(vb0[u] + dt * 16);
        vf.y = __builtin_nontemporal_load(vb1[u] + dt * 16);
        c = wmma_f32(vf, pf[u], c);
      }
      oacc[dt] = c;
    }
  }

  // ---- publish per-wave partials, then merge across waves ------------------
  if (lane < GRP) {                      // lanes 0..3 hold full stats for g
    wmS[wave * GRP + lane] = m;
    wlS[wave * GRP + lane] = l;
  }
  if (l16 < GRP) {                       // O^T: row d = dt*16 + j + 8*half, col g=l16
#pragma unroll
    for (int dt = 0; dt < 8; ++dt)
#pragma unroll
      for (int j = 0; j < 8; ++j)
        cacc[wave][l16 * DH + dt * 16 + j + 8 * half] = oacc[dt][j];
  }
  __syncthreads();

  {
    const int g = tid >> 7;              // 512 threads == 4*128 outputs
    const int d = tid & 127;
    float M = NEG_BIG;
    for (int w = 0; w < NW; ++w) M = fmaxf(M, wmS[w * GRP + g]);
    float num = 0.0f, den = 0.0f;
    for (int w = 0; w < NW; ++w) {
      const float wt = __expf(wmS[w * GRP + g] - M);
      num += wt * cacc[w][g * DH + d];
      den += wt * wlS[w * GRP + g];
    }
    out[((size_t)b * HQ + (size_t)(kvh * GRP + g)) * DH + d] = num / den;
  }
}

extern "C" void kernel_launch(void* const* d_in, const int* in_sizes, int n_in,
                              void* d_out, int out_size, void* d_ws, size_t ws_size,
                              hipStream_t stream) {
  (void)in_sizes; (void)n_in; (void)d_ws; (void)ws_size; (void)out_size;
  const float* q    = (const float*)d_in[0];
  const float* kc   = (const float*)d_in[1];
  const float* vc   = (const float*)d_in[2];
  const int*   lens = (const int*)d_in[3];
  const int*   bt   = (const int*)d_in[4];
  float* out = (float*)d_out;

  dim3 grid(BQ * KVHN);     // 128 blocks: one per (b, kv_head)
  dim3 block(NTHREADS);     // 16 wave32 waves
  gqa_flash_decode_kernel<<<grid, block, 0, stream>>>(q, kc, vc, lens, bt, out);
}